// FmInteraction_16432544874936
// MI455X (gfx1250) — compile-verified
//
#include <hip/hip_runtime.h>
#include <hip/hip_bf16.h>

typedef __attribute__((ext_vector_type(2))) float v2f;
typedef __attribute__((ext_vector_type(8))) float v8f;

__device__ __forceinline__ v8f wmma_f32_16x16x4(v2f a, v2f b, v8f c) {
    // 8 args: (neg_a, A, neg_b, B, c_mod, C, reuse_a, reuse_b)
    return __builtin_amdgcn_wmma_f32_16x16x4_f32(
        /*neg_a=*/false, a, /*neg_b=*/false, b,
        /*c_mod=*/(short)0, c, /*reuse_a=*/false, /*reuse_b=*/false);
}

// acc[64] += sum_i vals[i] * ff[ids[i], 0:64]
// WMMA formulation: per chunk of 4 ids, A(16x4)[f][k] = vals[i0+k]*ff[ids[i0+k]][fb+f],
// B(4x16) = ones  =>  C[f][n] += partial a[fb+f] in every column n.
// A-matrix 16x4 f32 layout (ISA 7.12.2): lane l -> M = l%16; VGPR0 holds K=2*(l/16),
// VGPR1 holds K=2*(l/16)+1. C/D layout: VGPR v, lanes 0-15 -> M=v, lanes 16-31 -> M=v+8.
__global__ void fm_weighted_sum_kernel(const int* __restrict__ ids,
                                       const float* __restrict__ vals,
                                       const float* __restrict__ ff,
                                       float* __restrict__ acc,
                                       int n_ids) {
    __shared__ float smem[64];
    const int tid = threadIdx.x;
    if (tid < 64) smem[tid] = 0.0f;
    __syncthreads();

    const int lane            = tid & 31;
    const int wave_in_block   = tid >> 5;
    const int waves_per_block = blockDim.x >> 5;
    const int global_wave     = blockIdx.x * waves_per_block + wave_in_block;
    const int total_waves     = gridDim.x * waves_per_block;

    const int m  = lane & 15;   // feature within 16-wide block
    const int kh = lane >> 4;   // which K pair this half-wave holds

    v8f c0 = {}, c1 = {}, c2 = {}, c3 = {};
    const v2f ones = {1.0f, 1.0f};

    const int nchunks = n_ids >> 2;  // n_ids is a multiple of 4 (4096 / 8192)
    for (int chunk = global_wave; chunk < nchunks; chunk += total_waves) {
        const int i0 = chunk * 4 + 2 * kh;
        const int i1 = i0 + 1;
        const int r0 = ids[i0];
        const int r1 = ids[i1];
        const float v0 = vals[i0];
        const float v1 = vals[i1];
        const float* p0 = ff + (size_t)r0 * 64 + m;
        const float* p1 = ff + (size_t)r1 * 64 + m;

        v2f a;
        a[0] = v0 * p0[0];  a[1] = v1 * p1[0];  c0 = wmma_f32_16x16x4(a, ones, c0);
        a[0] = v0 * p0[16]; a[1] = v1 * p1[16]; c1 = wmma_f32_16x16x4(a, ones, c1);
        a[0] = v0 * p0[32]; a[1] = v1 * p1[32]; c2 = wmma_f32_16x16x4(a, ones, c2);
        a[0] = v0 * p0[48]; a[1] = v1 * p1[48]; c3 = wmma_f32_16x16x4(a, ones, c3);
    }

    // Every column of each accumulator tile is identical; lanes with N==0
    // (lane 0 -> M=v, lane 16 -> M=v+8) publish their 8 rows per tile.
    if (m == 0) {
        const int base = kh * 8;
        #pragma unroll
        for (int v = 0; v < 8; ++v) {
            atomicAdd(&smem[ 0 + base + v], c0[v]);
            atomicAdd(&smem[16 + base + v], c1[v]);
            atomicAdd(&smem[32 + base + v], c2[v]);
            atomicAdd(&smem[48 + base + v], c3[v]);
        }
    }
    __syncthreads();
    if (tid < 64) atomicAdd(&acc[tid], smem[tid]);
}

__global__ void fm_zero_kernel(float* __restrict__ acc) {
    acc[threadIdx.x] = 0.0f;  // 128 threads zero a[64] + b[64]
}

__global__ void fm_dot_kernel(const float* __restrict__ acc, float* __restrict__ out) {
    __shared__ float s[64];
    const int l = threadIdx.x;        // 64 threads
    s[l] = acc[l] * acc[64 + l];
    __syncthreads();
    #pragma unroll
    for (int off = 32; off > 0; off >>= 1) {
        if (l < off) s[l] += s[l + off];
        __syncthreads();
    }
    if (l == 0) out[0] = s[0];
}

extern "C" void kernel_launch(void* const* d_in, const int* in_sizes, int n_in,
                              void* d_out, int out_size, void* d_ws, size_t ws_size,
                              hipStream_t stream) {
    const int*   left_ids   = (const int*)d_in[0];
    const float* left_vals  = (const float*)d_in[1];
    const int*   right_ids  = (const int*)d_in[2];
    const float* right_vals = (const float*)d_in[3];
    const float* lff        = (const float*)d_in[4];
    const float* rff        = (const float*)d_in[5];
    float* acc = (float*)d_ws;     // acc[0:64] = a, acc[64:128] = b
    float* out = (float*)d_out;

    const int n1 = in_sizes[0];    // 4096
    const int n2 = in_sizes[2];    // 8192

    fm_zero_kernel<<<1, 128, 0, stream>>>(acc);
    fm_weighted_sum_kernel<<<32, 256, 0, stream>>>(left_ids,  left_vals,  lff, acc,      n1);
    fm_weighted_sum_kernel<<<64, 256, 0, stream>>>(right_ids, right_vals, rff, acc + 64, n2);
    fm_dot_kernel<<<1, 64, 0, stream>>>(acc, out);
}